// MambaRecurrence_41386304864313
// MI455X (gfx1250) — compile-verified
//
#include <hip/hip_runtime.h>
#include <math.h>

// Mamba selective-scan for MI455X (gfx1250, wave32).
// Sizes fixed by the reference: B=2, L=2048, D=1024, N=16, R=64, E=96.

#define D_MODEL 1024
#define D_STATE 16
#define DT_RANK 64
#define E_DIM   96              // DT_RANK + 2*D_STATE
#define SEQ_L   2048
#define CHUNK   256             // L-chunk for the parallel scan
#define NCHUNK  (SEQ_L / CHUNK) // 8
#define LTILE   64              // timesteps staged in LDS at once

typedef __attribute__((ext_vector_type(2))) float v2f;
typedef __attribute__((ext_vector_type(8))) float v8f;

__device__ __forceinline__ v8f wmma_f32_k4(v2f a, v2f b, v8f c) {
  // D = A(16x4,f32) * B(4x16,f32) + C(16x16,f32), full f32 precision.
  return __builtin_amdgcn_wmma_f32_16x16x4_f32(false, a, false, b, (short)0, c,
                                               false, false);
}

// CDNA5 async global->LDS copy (ASYNCcnt path, no VGPR round-trip).
// Generic __shared__ pointers carry the LDS byte offset in their low 32 bits
// (aperture form), which is exactly what the VDST operand wants.
__device__ __forceinline__ void async_ld_b128(void* lds_dst, const void* gsrc) {
  unsigned ldsaddr = (unsigned)(size_t)lds_dst;
  asm volatile("global_load_async_to_lds_b128 %0, %1, off"
               :: "v"(ldsaddr), "v"(gsrc) : "memory");
}
__device__ __forceinline__ void async_wait0() {
  asm volatile("s_wait_asynccnt 0x0" ::: "memory");
}

// ---------------------------------------------------------------------------
// Kernel 1: x_dbl[m][e] = sum_d H[m][d] * Wx[e][d]     (M x 96, K = 1024)
// One wave computes a 16(M) x 96(N) strip; 8 waves / block.
// A-frag: lane<16 -> (M=lane, K=k..k+1), lane>=16 -> (M=lane-16, K=k+2..k+3).
// ---------------------------------------------------------------------------
__global__ __launch_bounds__(256) void xproj_gemm(const float* __restrict__ H,
                                                  const float* __restrict__ Wx,
                                                  float* __restrict__ xdbl,
                                                  int M) {
  const int lane  = threadIdx.x & 31;
  const int wave  = threadIdx.x >> 5;
  const int m0    = (blockIdx.x * 8 + wave) * 16;
  if (m0 >= M) return;                 // uniform per wave; EXEC stays all-ones
  const int row   = lane & 15;
  const int khalf = lane >> 4;         // 0 -> K0/K1, 1 -> K2/K3

  v8f acc[6] = {};
  for (int k = 0; k < D_MODEL; k += 4) {
    v2f a = *(const v2f*)(H + (size_t)(m0 + row) * D_MODEL + k + khalf * 2);
#pragma unroll
    for (int nt = 0; nt < 6; ++nt) {
      v2f b = *(const v2f*)(Wx + (size_t)(nt * 16 + row) * D_MODEL + k + khalf * 2);
      acc[nt] = wmma_f32_k4(a, b, acc[nt]);
    }
  }
  // D layout: VGPR r -> (M = m0 + khalf*8 + r, N = nt*16 + row)
  const int mb = m0 + khalf * 8;
#pragma unroll
  for (int nt = 0; nt < 6; ++nt)
#pragma unroll
    for (int r = 0; r < 8; ++r)
      xdbl[(size_t)(mb + r) * E_DIM + nt * 16 + row] = acc[nt][r];
}

// ---------------------------------------------------------------------------
// Kernel 2: delta[m][d] = softplus( sum_r xdbl[m][r] * Wdt[d][r] + b_dt[d] )
// M x 1024, K = 64. One wave computes 16(M) x 64(N); 8 waves / block.
// ---------------------------------------------------------------------------
__global__ __launch_bounds__(256) void dt_gemm(const float* __restrict__ xdbl,
                                               const float* __restrict__ Wdt,
                                               const float* __restrict__ bdt,
                                               float* __restrict__ delta,
                                               int M) {
  const int lane  = threadIdx.x & 31;
  const int wave  = threadIdx.x >> 5;
  const int tiles_n = D_MODEL / 64;    // 16
  const int wt  = blockIdx.x * 8 + wave;
  const int m0  = (wt / tiles_n) * 16;
  const int n0  = (wt % tiles_n) * 64;
  if (m0 >= M) return;
  const int row   = lane & 15;
  const int khalf = lane >> 4;

  v8f acc[4] = {};
  for (int k = 0; k < DT_RANK; k += 4) {
    v2f a = *(const v2f*)(xdbl + (size_t)(m0 + row) * E_DIM + k + khalf * 2);
#pragma unroll
    for (int nt = 0; nt < 4; ++nt) {
      v2f b = *(const v2f*)(Wdt + (size_t)(n0 + nt * 16 + row) * DT_RANK + k + khalf * 2);
      acc[nt] = wmma_f32_k4(a, b, acc[nt]);
    }
  }
  const int mb = m0 + khalf * 8;
#pragma unroll
  for (int nt = 0; nt < 4; ++nt)
#pragma unroll
    for (int r = 0; r < 8; ++r) {
      const int d = n0 + nt * 16 + row;
      float z = acc[nt][r] + bdt[d];
      float sp = (z > 20.f) ? z : log1pf(__expf(z));   // stable softplus
      delta[(size_t)(mb + r) * D_MODEL + d] = sp;
    }
}

// ---------------------------------------------------------------------------
// Scan phase A: per (b, chunk, d, n) compute the chunk transfer function:
//   aprod = prod_l exp(delta*A),  hloc = local scan end-state with h0 = 0.
// Block = 256 threads = 16 d-channels x 16 states.
// delta/x/B tiles staged via CDNA5 async global->LDS (one b128 per thread).
// ---------------------------------------------------------------------------
__global__ __launch_bounds__(256) void scan_phaseA(const float* __restrict__ H,
                                                   const float* __restrict__ xdbl,
                                                   const float* __restrict__ delta,
                                                   const float* __restrict__ Alog,
                                                   float* __restrict__ aprod_ws,
                                                   float* __restrict__ hloc_ws) {
  const int DTILES = D_MODEL / 16;     // 64
  int bid   = blockIdx.x;
  const int dtile = bid % DTILES; bid /= DTILES;
  const int ch    = bid % NCHUNK; bid /= NCHUNK;
  const int b     = bid;

  const int t  = threadIdx.x;
  const int dl = t >> 4;               // 0..15 : d within tile
  const int n  = t & 15;               // 0..15 : state index
  const int d  = dtile * 16 + dl;

  const float a = -__expf(Alog[d * D_STATE + n]);

  __shared__ __align__(16) float sDel[LTILE * 16];
  __shared__ __align__(16) float sX  [LTILE * 16];
  __shared__ __align__(16) float sB  [LTILE * 16];

  // staging map: thread t moves one float4: row lr = t/4, cols c4..c4+3
  const int lr = t >> 2;
  const int c4 = (t & 3) * 4;

  float h = 0.f, ap = 1.f;
  for (int l0 = 0; l0 < CHUNK; l0 += LTILE) {
    const int lbase = ch * CHUNK + l0;
    __syncthreads();                                   // readers of prev tile done
    {
      const size_t g = (size_t)b * SEQ_L + lbase + lr;
      async_ld_b128(&sDel[lr * 16 + c4], delta + g * D_MODEL + dtile * 16 + c4);
      async_ld_b128(&sX[lr * 16 + c4],   H     + g * D_MODEL + dtile * 16 + c4);
      async_ld_b128(&sB[lr * 16 + c4],   xdbl  + g * E_DIM + DT_RANK + c4);
    }
    async_wait0();                                     // this wave's asyncs landed
    __syncthreads();                                   // all waves' asyncs landed
#pragma unroll 4
    for (int li = 0; li < LTILE; ++li) {
      const float dlt = sDel[li * 16 + dl];
      const float x   = sX[li * 16 + dl];
      const float Bv  = sB[li * 16 + n];
      const float da  = __expf(dlt * a);
      h  = da * h + (dlt * x) * Bv;
      ap *= da;
    }
  }
  const size_t idx = (((size_t)b * NCHUNK + ch) * D_MODEL + d) * D_STATE + n;
  aprod_ws[idx] = ap;
  hloc_ws[idx]  = h;
}

// ---------------------------------------------------------------------------
// Scan phase B: tiny sequential combine across the 8 chunks per (b,d,n).
//   hinit[c] = state entering chunk c.
// ---------------------------------------------------------------------------
__global__ void scan_phaseB(const float* __restrict__ aprod,
                            const float* __restrict__ hloc,
                            float* __restrict__ hinit, int total) {
  const int idx = blockIdx.x * blockDim.x + threadIdx.x;   // over B*D*N
  if (idx >= total) return;
  const int b  = idx / (D_MODEL * D_STATE);
  const int dn = idx % (D_MODEL * D_STATE);
  float h = 0.f;
  for (int c = 0; c < NCHUNK; ++c) {
    const size_t g = ((size_t)b * NCHUNK + c) * (D_MODEL * D_STATE) + dn;
    hinit[g] = h;
    h = aprod[g] * h + hloc[g];
  }
}

// ---------------------------------------------------------------------------
// Scan phase C: re-run each chunk with its correct initial state and emit
//   y[b,l,d] = sum_n h[n] * C[n]   (16-wide shfl_xor reduction over states).
// ---------------------------------------------------------------------------
__global__ __launch_bounds__(256) void scan_phaseC(const float* __restrict__ H,
                                                   const float* __restrict__ xdbl,
                                                   const float* __restrict__ delta,
                                                   const float* __restrict__ Alog,
                                                   const float* __restrict__ hinit,
                                                   float* __restrict__ out) {
  const int DTILES = D_MODEL / 16;
  int bid   = blockIdx.x;
  const int dtile = bid % DTILES; bid /= DTILES;
  const int ch    = bid % NCHUNK; bid /= NCHUNK;
  const int b     = bid;

  const int t  = threadIdx.x;
  const int dl = t >> 4;
  const int n  = t & 15;
  const int d  = dtile * 16 + dl;

  const float a = -__expf(Alog[d * D_STATE + n]);
  float h = hinit[(((size_t)b * NCHUNK + ch) * D_MODEL + d) * D_STATE + n];

  __shared__ __align__(16) float sDel[LTILE * 16];
  __shared__ __align__(16) float sX  [LTILE * 16];
  __shared__ __align__(16) float sB  [LTILE * 16];
  __shared__ __align__(16) float sC  [LTILE * 16];

  const int lr = t >> 2;
  const int c4 = (t & 3) * 4;

  for (int l0 = 0; l0 < CHUNK; l0 += LTILE) {
    const int lbase = ch * CHUNK + l0;
    __syncthreads();
    {
      const size_t g = (size_t)b * SEQ_L + lbase + lr;
      async_ld_b128(&sDel[lr * 16 + c4], delta + g * D_MODEL + dtile * 16 + c4);
      async_ld_b128(&sX[lr * 16 + c4],   H     + g * D_MODEL + dtile * 16 + c4);
      async_ld_b128(&sB[lr * 16 + c4],   xdbl  + g * E_DIM + DT_RANK + c4);
      async_ld_b128(&sC[lr * 16 + c4],   xdbl  + g * E_DIM + DT_RANK + D_STATE + c4);
    }
    async_wait0();
    __syncthreads();
    for (int li = 0; li < LTILE; ++li) {
      const float dlt = sDel[li * 16 + dl];
      const float x   = sX[li * 16 + dl];
      const float Bv  = sB[li * 16 + n];
      const float Cv  = sC[li * 16 + n];
      const float da  = __expf(dlt * a);
      h = da * h + (dlt * x) * Bv;
      float y = h * Cv;
      // reduce over the 16 states (lanes n=0..15 within each 16-lane group)
      y += __shfl_xor(y, 1, 16);
      y += __shfl_xor(y, 2, 16);
      y += __shfl_xor(y, 4, 16);
      y += __shfl_xor(y, 8, 16);
      if (n == 0)
        out[((size_t)b * SEQ_L + lbase + li) * D_MODEL + d] = y;
    }
  }
}

// ---------------------------------------------------------------------------
extern "C" void kernel_launch(void* const* d_in, const int* in_sizes, int n_in,
                              void* d_out, int out_size, void* d_ws, size_t ws_size,
                              hipStream_t stream) {
  const float* H    = (const float*)d_in[0];   // (B, L, D)
  const float* Wx   = (const float*)d_in[1];   // (96, D)
  const float* Wdt  = (const float*)d_in[2];   // (D, 64)
  const float* bdt  = (const float*)d_in[3];   // (D,)
  const float* Alog = (const float*)d_in[4];   // (D, 16)
  float* out = (float*)d_out;

  const int M  = in_sizes[0] / D_MODEL;        // B*L = 4096
  const int Bb = M / SEQ_L;                    // 2

  char* ws = (char*)d_ws;
  float* xdbl  = (float*)ws;  ws += (size_t)M * E_DIM   * sizeof(float);
  float* delta = (float*)ws;  ws += (size_t)M * D_MODEL * sizeof(float);
  const size_t cs = (size_t)Bb * NCHUNK * D_MODEL * D_STATE * sizeof(float);
  float* aprod = (float*)ws;  ws += cs;
  float* hloc  = (float*)ws;  ws += cs;
  float* hinit = (float*)ws;

  xproj_gemm<<<M / 128, 256, 0, stream>>>(H, Wx, xdbl, M);
  dt_gemm<<<(M / 16) * (D_MODEL / 64) / 8, 256, 0, stream>>>(xdbl, Wdt, bdt, delta, M);
  scan_phaseA<<<Bb * NCHUNK * (D_MODEL / 16), 256, 0, stream>>>(H, xdbl, delta, Alog,
                                                                aprod, hloc);
  const int totBDN = Bb * D_MODEL * D_STATE;
  scan_phaseB<<<(totBDN + 255) / 256, 256, 0, stream>>>(aprod, hloc, hinit, totBDN);
  scan_phaseC<<<Bb * NCHUNK * (D_MODEL / 16), 256, 0, stream>>>(H, xdbl, delta, Alog,
                                                                hinit, out);
}